// Kipfblock_78039555768469
// MI455X (gfx1250) — compile-verified
//
#include <hip/hip_runtime.h>
#include <hip/hip_bf16.h>

typedef __attribute__((ext_vector_type(2))) float v2f;
typedef __attribute__((ext_vector_type(8))) float v8f;

#define F 64  // F_IN == F_HID == 64

// ---------------------------------------------------------------------------
// out[i] = (beta==0) ? 0 : beta*prev[i]
__global__ __launch_bounds__(256) void init_buf(float* __restrict__ out,
                                                const float* __restrict__ prev,
                                                float beta, int n) {
    int i = blockIdx.x * blockDim.x + threadIdx.x;
    if (i < n) out[i] = (beta == 0.0f) ? 0.0f : beta * prev[i];
}

// deg[row[e]] += 1  (counts < 2^24 so f32 atomic is exact)
__global__ __launch_bounds__(256) void deg_kernel(const int* __restrict__ row,
                                                  float* __restrict__ deg, int E) {
    int e = blockIdx.x * blockDim.x + threadIdx.x;
    if (e < E) atomicAdd(&deg[row[e]], 1.0f);
}

// in-place deg -> dinv
__global__ __launch_bounds__(256) void dinv_kernel(float* __restrict__ d, int n) {
    int i = blockIdx.x * blockDim.x + threadIdx.x;
    if (i < n) {
        float v = d[i];
        d[i] = (v > 0.0f) ? rsqrtf(v) : 0.0f;
    }
}

// w[e] = -(dinv[row]*dinv[col])
__global__ __launch_bounds__(256) void w_kernel(const int* __restrict__ row,
                                                const int* __restrict__ col,
                                                const float* __restrict__ dinv,
                                                float* __restrict__ w, int E) {
    int e = blockIdx.x * blockDim.x + threadIdx.x;
    if (e < E) w[e] = -(dinv[row[e]] * dinv[col[e]]);
}

// Scatter: out[col[e]] += alpha*w[e]*h[row[e]].  16 lanes per edge, float4 each.
// One coalesced 256B gather per edge; 4 f32 atomics per lane resolved in L2.
__global__ __launch_bounds__(256) void prop_scatter(const int* __restrict__ row,
                                                    const int* __restrict__ col,
                                                    const float* __restrict__ w,
                                                    const float* __restrict__ h,
                                                    float* __restrict__ out,
                                                    float alpha, int E) {
    long tid = (long)blockIdx.x * blockDim.x + threadIdx.x;
    int e  = (int)(tid >> 4);
    int c4 = (int)(tid & 15) << 2;
    if (e >= E) return;
    int r = row[e];
    int d = col[e];
    float wv = alpha * w[e];
    const float4 v = *reinterpret_cast<const float4*>(h + (long)r * F + c4);
    float* o = out + (long)d * F + c4;
    atomicAdd(o + 0, wv * v.x);
    atomicAdd(o + 1, wv * v.y);
    atomicAdd(o + 2, wv * v.z);
    atomicAdd(o + 3, wv * v.w);
}

// ---------------------------------------------------------------------------
// out[n_rows x 64] (+)= A[n_rows x 64] @ W[64 x 64] via V_WMMA_F32_16X16X4_F32.
// One wave computes a full 16x64 strip: 4 accumulators, one A frag feeds 4
// WMMAs per k-step (64 WMMAs per wave).  W_k is staged in LDS per block.
// A frag (16x4 f32): lane m=lane&15, kb=(lane>>4)*2 -> {A[m][kk+kb], A[m][kk+kb+1]}
// B frag (4x16 f32): lane n=lane&15                 -> {W[kk+kb][n], W[kk+kb+1][n]}
// C/D (16x16 f32):   VGPR r -> M = r + 8*(lane>>4), N = lane&15
__global__ __launch_bounds__(256) void gemm_cheb(const float* __restrict__ A,
                                                 const float* __restrict__ W,
                                                 float* __restrict__ out,
                                                 const float* __restrict__ bias,
                                                 int n_rows, int first, int last) {
    __shared__ float Wl[F * F];   // 16 KB of the WGP's 320 KB LDS

    // Cooperative stage of W_k: 1024 float4 across 256 threads.
    {
        const float4* src = reinterpret_cast<const float4*>(W);
        float4*       dst = reinterpret_cast<float4*>(Wl);
#pragma unroll
        for (int i = threadIdx.x; i < (F * F) / 4; i += 256)
            dst[i] = src[i];
    }
    __syncthreads();

    const int lane = threadIdx.x & 31;
    const int wave = threadIdx.x >> 5;
    const int row_tile = blockIdx.x * 8 + wave;   // block of 16 rows
    if (row_tile * 16 >= n_rows) return;          // wave-uniform exit (after barrier)

    const int m    = lane & 15;
    const int half = lane >> 4;
    const int kb   = half << 1;
    const int row0 = row_tile * 16;

    v8f c[4] = {};
    if (!first) {
#pragma unroll
        for (int t = 0; t < 4; ++t)
#pragma unroll
            for (int r = 0; r < 8; ++r)
                c[t][r] = out[(long)(row0 + r + 8 * half) * F + t * 16 + m];
    }

#pragma unroll
    for (int kk = 0; kk < F; kk += 4) {
        v2f a;
        a.x = A[(long)(row0 + m) * F + kk + kb];
        a.y = A[(long)(row0 + m) * F + kk + kb + 1];
#pragma unroll
        for (int t = 0; t < 4; ++t) {
            v2f b;
            b.x = Wl[(kk + kb) * F + t * 16 + m];
            b.y = Wl[(kk + kb + 1) * F + t * 16 + m];
            c[t] = __builtin_amdgcn_wmma_f32_16x16x4_f32(false, a, false, b,
                                                         (short)0, c[t],
                                                         false, false);
        }
    }

    if (last) {
#pragma unroll
        for (int t = 0; t < 4; ++t) {
            float bv = bias[t * 16 + m];
#pragma unroll
            for (int r = 0; r < 8; ++r) {
                float v = c[t][r] + bv;
                out[(long)(row0 + r + 8 * half) * F + t * 16 + m] =
                    v > 0.0f ? v : 0.0f;
            }
        }
    } else {
#pragma unroll
        for (int t = 0; t < 4; ++t)
#pragma unroll
            for (int r = 0; r < 8; ++r)
                out[(long)(row0 + r + 8 * half) * F + t * 16 + m] = c[t][r];
    }
}

// ---------------------------------------------------------------------------
extern "C" void kernel_launch(void* const* d_in, const int* in_sizes, int n_in,
                              void* d_out, int out_size, void* d_ws, size_t ws_size,
                              hipStream_t stream) {
    const float* x    = (const float*)d_in[0];
    const int*   ei   = (const int*)d_in[1];
    const float* Wmat = (const float*)d_in[2];
    const float* bias = (const float*)d_in[3];
    float*       out  = (float*)d_out;

    const int N = in_sizes[0] / F;
    const int E = in_sizes[1] / 2;
    const int K = in_sizes[2] / (F * F);
    const int* row = ei;       // edge_index[0] = source j
    const int* col = ei + E;   // edge_index[1] = target i

    // Workspace carving (256B aligned): dinv | w | T0 | T1 | T2  (~84 MB)
    char* ws = (char*)d_ws;
    size_t off = 0;
    auto carve = [&](size_t bytes) -> float* {
        float* p = (float*)(ws + off);
        off = (off + bytes + 255) & ~(size_t)255;
        return p;
    };
    float* dinv = carve((size_t)N * sizeof(float));
    float* wbuf = carve((size_t)E * sizeof(float));
    float* T[3];
    for (int i = 0; i < 3; ++i) T[i] = carve((size_t)N * F * sizeof(float));

    const int NF      = N * F;
    const int nbN     = (N + 255) / 256;
    const int nbE     = (E + 255) / 256;
    const int nbNF    = (NF + 255) / 256;
    const int pblocks = (int)(((long)E * 16 + 255) / 256);
    const int gblocks = ((N + 15) / 16 + 7) / 8;   // one wave per 16-row strip

    // 1) degree -> dinv -> edge weights
    init_buf<<<nbN, 256, 0, stream>>>(dinv, dinv, 0.0f, N);
    deg_kernel<<<nbE, 256, 0, stream>>>(row, dinv, E);
    dinv_kernel<<<nbN, 256, 0, stream>>>(dinv, N);
    w_kernel<<<nbE, 256, 0, stream>>>(row, col, dinv, wbuf, E);

    // 2) k = 0 : out = x @ W0
    gemm_cheb<<<gblocks, 256, 0, stream>>>(x, Wmat, out, bias, N, 1, K == 1);
    if (K == 1) return;

    // 3) k = 1 : T1 = prop(x); out += T1 @ W1
    const float* Tx0 = x;
    float* Tx1 = T[0];
    init_buf<<<nbNF, 256, 0, stream>>>(Tx1, Tx1, 0.0f, NF);
    prop_scatter<<<pblocks, 256, 0, stream>>>(row, col, wbuf, x, Tx1, 1.0f, E);
    gemm_cheb<<<gblocks, 256, 0, stream>>>(Tx1, Wmat + (size_t)F * F, out, bias,
                                           N, 0, K == 2);

    // 4) k >= 2 : T2 = 2*prop(T1) - T0  (init to -T0, scatter with weight 2w)
    int buf = 1;
    for (int k = 2; k < K; ++k) {
        float* Tx2 = T[buf];
        init_buf<<<nbNF, 256, 0, stream>>>(Tx2, Tx0, -1.0f, NF);
        prop_scatter<<<pblocks, 256, 0, stream>>>(row, col, wbuf, Tx1, Tx2, 2.0f, E);
        gemm_cheb<<<gblocks, 256, 0, stream>>>(Tx2, Wmat + (size_t)k * F * F, out,
                                               bias, N, 0, k == K - 1);
        Tx0 = Tx1;
        Tx1 = Tx2;
        buf = (buf + 1) % 3;
    }
}